// InhibitionLayer_56538949485246
// MI455X (gfx1250) — compile-verified
//
#include <hip/hip_runtime.h>
#include <hip/hip_bf16.h>

// Problem constants from the reference
#define N_INPUTS 16384
#define KDET 32
#define DET_PER_THREAD 8
#define THREADS 256

// CDNA5 async global->LDS staging path (ASYNCcnt-tracked), guarded so the
// file still compiles (with VGPR staging fallback) if the toolchain lacks it.
#if __has_builtin(__builtin_amdgcn_global_load_async_to_lds_b128) && \
    __has_builtin(__builtin_amdgcn_s_wait_asynccnt)
#define WTA_HAVE_ASYNC_LDS 1
#else
#define WTA_HAVE_ASYNC_LDS 0
#endif

#if WTA_HAVE_ASYNC_LDS
// Builtin signature (from hipcc diagnostic): param0 is
// 'int __attribute__((vector_size(16))) __device__ *' i.e. AS1 v4i*.
typedef int wta_v4i __attribute__((vector_size(16)));
typedef __attribute__((address_space(1))) wta_v4i wta_gv4i;  // global
typedef __attribute__((address_space(3))) wta_v4i wta_lv4i;  // LDS
#endif

// Pre-fill output with 1.0f ("spike unless inhibited")
__global__ __launch_bounds__(256)
void wta_fill_ones(float4* __restrict__ out, int n4) {
    int i = blockIdx.x * blockDim.x + threadIdx.x;
    if (i < n4) out[i] = make_float4(1.0f, 1.0f, 1.0f, 1.0f);
}

// One block = one batch row (x row staged in 64KB LDS) x 2048 detectors.
// Each thread: 8 detectors; per detector gather 32 values from LDS,
// find FIRST max slot (strict > over ascending k == jnp.argmax semantics),
// then store 0.0f to every other valid member's output slot.
__global__ __launch_bounds__(THREADS)
void wta_inhibit_kernel(const float* __restrict__ x,
                        const int*   __restrict__ det,
                        float*       __restrict__ out,
                        int nDet) {
    extern __shared__ float xs[];   // N_INPUTS floats = 64 KB

    const int b = blockIdx.y;
    const float* __restrict__ xrow = x + (size_t)b * N_INPUTS;

    // Stage this batch row into LDS.
    {
        const float4* __restrict__ x4 = reinterpret_cast<const float4*>(xrow);
        float4* __restrict__ xs4 = reinterpret_cast<float4*>(xs);
#if WTA_HAVE_ASYNC_LDS
        // global_load_async_to_lds_b128: 16B per lane per issue, no VGPR
        // round-trip; all 16 transfers per thread in flight on ASYNCcnt.
        #pragma unroll
        for (int i = threadIdx.x; i < N_INPUTS / 4; i += THREADS) {
            __builtin_amdgcn_global_load_async_to_lds_b128(
                (wta_gv4i*)(x4 + i),
                (wta_lv4i*)(xs4 + i),
                /*imm offset*/ 0, /*cpol*/ 0);
        }
        __builtin_amdgcn_s_wait_asynccnt(0);  // LDS writes done before barrier
#else
        #pragma unroll
        for (int i = threadIdx.x; i < N_INPUTS / 4; i += THREADS) {
            xs4[i] = x4[i];
        }
#endif
    }
    __syncthreads();

    float* __restrict__ outrow = out + (size_t)b * N_INPUTS;

    const int detPerBlock = THREADS * DET_PER_THREAD;
    const int d0 = blockIdx.x * detPerBlock + threadIdx.x;

    #pragma unroll
    for (int it = 0; it < DET_PER_THREAD; ++it) {
        const int d = d0 + it * THREADS;
        if (d >= nDet) continue;

        // Prefetch next detector row this thread will touch (global_prefetch_b8)
        if (it + 1 < DET_PER_THREAD) {
            const int dn = d0 + (it + 1) * THREADS;
            if (dn < nDet) {
                __builtin_prefetch(det + (size_t)dn * KDET, 0, 0);
            }
        }

        // Load 32 member ids (128B row) as 8 x int4, fully into registers
        const int4* __restrict__ r = reinterpret_cast<const int4*>(det + (size_t)d * KDET);
        int idx[KDET];
        #pragma unroll
        for (int j = 0; j < KDET / 4; ++j) {
            int4 q = r[j];
            idx[4 * j + 0] = q.x;
            idx[4 * j + 1] = q.y;
            idx[4 * j + 2] = q.z;
            idx[4 * j + 3] = q.w;
        }

        // Pass 1: first-occurrence argmax over valid members (ds_load_b32 gathers)
        float best = -__builtin_inff();
        int wslot = -1;
        #pragma unroll
        for (int k = 0; k < KDET; ++k) {
            const int id = idx[k];
            if (id >= 0) {
                const float v = xs[id];
                if (v > best) { best = v; wslot = k; }
            }
        }

        // Pass 2: inhibit every valid non-winner slot (benign-race 0.0f stores)
        #pragma unroll
        for (int k = 0; k < KDET; ++k) {
            const int id = idx[k];
            if (id >= 0 && k != wslot) {
                outrow[id] = 0.0f;
            }
        }
    }
}

extern "C" void kernel_launch(void* const* d_in, const int* in_sizes, int n_in,
                              void* d_out, int out_size, void* d_ws, size_t ws_size,
                              hipStream_t stream) {
    (void)d_ws; (void)ws_size; (void)n_in;

    const float* x   = (const float*)d_in[0];
    const int*   det = (const int*)d_in[1];
    float*       out = (float*)d_out;

    const int B    = in_sizes[0] / N_INPUTS;   // 32
    const int nDet = in_sizes[1] / KDET;       // 65536

    // 1) out = 1.0f everywhere
    {
        const int n4 = out_size / 4;
        const int blocks = (n4 + 255) / 256;
        wta_fill_ones<<<blocks, 256, 0, stream>>>((float4*)out, n4);
    }

    // 2) scatter 0.0f to inhibited inputs
    {
        const int detPerBlock = THREADS * DET_PER_THREAD;       // 2048
        dim3 grid((nDet + detPerBlock - 1) / detPerBlock, B);   // (32, 32)
        const size_t ldsBytes = (size_t)N_INPUTS * sizeof(float); // 64 KB
        wta_inhibit_kernel<<<grid, THREADS, ldsBytes, stream>>>(x, det, out, nDet);
    }
}